// MHCResidual_33251636805747
// MI455X (gfx1250) — compile-verified
//
#include <hip/hip_runtime.h>
#include <hip/hip_bf16.h>

// ---------------------------------------------------------------------------
// MHC residual block, fused single-pass kernel for MI455X (gfx1250, wave32).
//
// Shapes: B=4 S=2048 n=4 C=1024  -> T=8192 tokens, nC=4096 per token.
// Memory-bound (256 MB min traffic, ~11us @ 23.3 TB/s). One workgroup per
// 16 tokens: stage 16x4096 fp32 tile in LDS (~287 KB incl. pad, < 320 KB WGP
// LDS), do RMS + projection GEMM (V_WMMA_F32_16X16X4_F32) + sinkhorn + mix
// entirely on-chip, single read of x, single write of out.
//
// Staging uses GLOBAL_LOAD_ASYNC_TO_LDS_B128 (ASYNCcnt) when the toolchain
// exposes the builtin; otherwise falls back to b128 loads + ds stores.
// ---------------------------------------------------------------------------

typedef __attribute__((ext_vector_type(2))) float v2f;
typedef __attribute__((ext_vector_type(8))) float v8f;
typedef int v4i_vs __attribute__((vector_size(16)));   // matches builtin proto
typedef __attribute__((address_space(1))) v4i_vs* g_v4i_p;
typedef __attribute__((address_space(3))) v4i_vs* l_v4i_p;

#define TOKENS_PER_WG 16
#define NC           4096
#define PITCH        4100      // 4096 + 4 floats pad -> conflict-free A reads
#define NWAVE        8         // 256 threads
#define KG_TOTAL     1024      // 4096 / 4 k-groups
#define KG_PER_WAVE  (KG_TOTAL / NWAVE)

// LDS layout (floats)
#define XS_F     (TOKENS_PER_WG * PITCH)   // 65600
#define RED_F    (NWAVE * 2 * 8 * 32)      // 4096  wave-partial C tiles
#define PS_F     (TOKENS_PER_WG * 32)      // 512   p logits (padded to 32)
#define WSF      1024                      // rms_weight
#define RINV_F   TOKENS_PER_WG
#define HPRE_F   (TOKENS_PER_WG * 4)
#define HPOST_F  (TOKENS_PER_WG * 4)
#define MS_F     (TOKENS_PER_WG * 16)
#define LDS_FLOATS (XS_F + RED_F + PS_F + WSF + RINV_F + HPRE_F + HPOST_F + MS_F)

#if defined(__has_builtin)
#if __has_builtin(__builtin_amdgcn_global_load_async_to_lds_b128)
#define HAS_ASYNC_LDS 1
#endif
#endif
#ifndef HAS_ASYNC_LDS
#define HAS_ASYNC_LDS 0
#endif

// ---------------------------------------------------------------------------
// Setup: swizzle phi (24 x 4096, padded to 32 rows) into the per-lane WMMA
// B-matrix layout so the hot loop's B loads are coalesced float2 per lane.
// Layout: Bbuf[(tile*1024 + g)*32 + lane] = {B(k, n), B(k+1, n)} with
//   n = tile*16 + (lane & 15),  k = g*4 + (lane >= 16 ? 2 : 0)
// matching A layout VGPR0 = K0|K2, VGPR1 = K1|K3.
// ---------------------------------------------------------------------------
__global__ __launch_bounds__(256) void build_Bbuf(
    const float* __restrict__ phi_pre, const float* __restrict__ phi_post,
    const float* __restrict__ phi_res, v2f* __restrict__ Bbuf)
{
    int idx  = blockIdx.x * 256 + threadIdx.x;      // 0 .. 65535
    int lane = idx & 31;
    int g    = (idx >> 5) & 1023;
    int tile = idx >> 15;                            // 0 or 1
    int n    = tile * 16 + (lane & 15);
    int k    = g * 4 + ((lane >= 16) ? 2 : 0);

    float vx, vy;
    if (n < 4)       { vx = phi_pre [(size_t)n        * NC + k];
                       vy = phi_pre [(size_t)n        * NC + k + 1]; }
    else if (n < 8)  { vx = phi_post[(size_t)(n - 4)  * NC + k];
                       vy = phi_post[(size_t)(n - 4)  * NC + k + 1]; }
    else if (n < 24) { vx = phi_res [(size_t)(n - 8)  * NC + k];
                       vy = phi_res [(size_t)(n - 8)  * NC + k + 1]; }
    else             { vx = 0.f; vy = 0.f; }

    v2f b; b.x = vx; b.y = vy;
    Bbuf[idx] = b;
}

// ---------------------------------------------------------------------------
// Main fused kernel: 512 blocks x 256 threads (8 waves), dynamic LDS.
// ---------------------------------------------------------------------------
__global__ __launch_bounds__(256) void mhc_fused(
    const float* __restrict__ x,        // (T, 4, 1024) viewed as (T, 4096)
    const v2f*   __restrict__ Bbuf,     // swizzled phi
    const float* __restrict__ b_pre,    // (4)
    const float* __restrict__ b_post,   // (4)
    const float* __restrict__ b_res,    // (16)
    const float* __restrict__ alpha_pre,
    const float* __restrict__ alpha_post,
    const float* __restrict__ alpha_res,
    const float* __restrict__ rms_w,    // (1024)
    float*       __restrict__ out)      // (T, 4, 1024)
{
    extern __shared__ float lds[];
    float* xs     = lds;                 // [16][4100]
    float* red    = xs     + XS_F;       // [8 waves][2 tiles][8 v][32 lanes]
    float* p_s    = red    + RED_F;      // [16][32]
    float* w_s    = p_s    + PS_F;       // [1024]
    float* rinv_s = w_s    + WSF;        // [16]
    float* Hpre_s = rinv_s + RINV_F;     // [16][4]
    float* Hpost_s= Hpre_s + HPRE_F;     // [16][4]
    float* M_s    = Hpost_s+ HPOST_F;    // [16][16]

    const int tid  = threadIdx.x;
    const int lane = tid & 31;
    const int wave = tid >> 5;
    const long tok0 = (long)blockIdx.x * TOKENS_PER_WG;

    // ---- 1) stage x tile into LDS, stage rms_weight ----
    for (int base = tid * 4; base < TOKENS_PER_WG * NC; base += 256 * 4) {
        int t = base >> 12;              // / 4096
        int c = base & (NC - 1);
#if HAS_ASYNC_LDS
        // direct global -> LDS, 16B per lane, tracked by ASYNCcnt
        __builtin_amdgcn_global_load_async_to_lds_b128(
            (g_v4i_p)(x + (tok0 + t) * (size_t)NC + c),
            (l_v4i_p)(xs + t * PITCH + c),
            0, 0);
#else
        float4 v = *(const float4*)(x + (tok0 + t) * (size_t)NC + c);
        *(float4*)(xs + t * PITCH + c) = v;
#endif
    }
    for (int c = tid; c < 1024; c += 256) w_s[c] = rms_w[c];
#if HAS_ASYNC_LDS
    asm volatile("s_wait_asynccnt 0x0" ::: "memory");
#endif
    __syncthreads();

    // ---- 2) per-token sumsq -> 1/rms (2 tokens per wave, float4 reads) ----
    for (int tt = 0; tt < 2; ++tt) {
        int t = wave + tt * NWAVE;
        float ss = 0.f;
        const float* xb = xs + t * PITCH;
        for (int c4 = lane; c4 < NC / 4; c4 += 32) {
            float4 v = *(const float4*)(xb + c4 * 4);
            ss += v.x * v.x + v.y * v.y + v.z * v.z + v.w * v.w;
        }
        for (int off = 16; off; off >>= 1) ss += __shfl_xor(ss, off, 32);
        if (lane == 0) rinv_s[t] = rsqrtf(ss * (1.0f / NC) + 1e-6f);
    }

    // ---- 3) projection GEMM via V_WMMA_F32_16X16X4_F32 ----
    // A: 16 tokens x 4 K from LDS; B: two 16-col tiles from swizzled phi.
    // Wave w covers k-groups [w*128, w*128+128). Uniform control flow: EXEC=~0.
    v8f acc0 = {0.f,0.f,0.f,0.f,0.f,0.f,0.f,0.f};
    v8f acc1 = {0.f,0.f,0.f,0.f,0.f,0.f,0.f,0.f};
    {
        const int tA    = lane & 15;
        const int koffA = (lane >= 16) ? 2 : 0;
        const float* aBase = xs + tA * PITCH + koffA;
        const v2f* b0p = Bbuf + lane;                    // tile 0
        const v2f* b1p = Bbuf + (size_t)1024 * 32 + lane; // tile 1
        const int g0 = wave * KG_PER_WAVE;
        #pragma unroll 4
        for (int g = g0; g < g0 + KG_PER_WAVE; ++g) {
            v2f a  = *(const v2f*)(aBase + g * 4);        // ds_load_b64
            v2f b0 = b0p[(size_t)g * 32];                 // coalesced global b64
            v2f b1 = b1p[(size_t)g * 32];
            acc0 = __builtin_amdgcn_wmma_f32_16x16x4_f32(
                       false, a, false, b0, (short)0, acc0, false, false);
            acc1 = __builtin_amdgcn_wmma_f32_16x16x4_f32(
                       false, a, false, b1, (short)0, acc1, false, false);
        }
    }
    // stash wave partials
    #pragma unroll
    for (int v = 0; v < 8; ++v) {
        red[((wave * 2 + 0) * 8 + v) * 32 + lane] = acc0[v];
        red[((wave * 2 + 1) * 8 + v) * 32 + lane] = acc1[v];
    }
    __syncthreads();

    // cross-wave reduce 2x8x32 C entries; map (tile,v,lane) -> (token, n)
    for (int i = tid; i < 2 * 8 * 32; i += 256) {
        int tile = i >> 8, v = (i >> 5) & 7, l = i & 31;
        float s = 0.f;
        #pragma unroll
        for (int wv = 0; wv < NWAVE; ++wv)
            s += red[((wv * 2 + tile) * 8 + v) * 32 + l];
        int token = v + ((l >= 16) ? 8 : 0);
        int n     = tile * 16 + (l & 15);
        p_s[token * 32 + n] = s;
    }
    __syncthreads();

    // ---- 4) per-token epilogue: gates + sinkhorn (one thread per token) ----
    if (tid < TOKENS_PER_WG) {
        const int t = tid;
        const float rinv = rinv_s[t];
        const float ap = alpha_pre[0], aq = alpha_post[0], ar = alpha_res[0];
        float m[16];
        #pragma unroll
        for (int i = 0; i < 4; ++i) {
            float pp = p_s[t * 32 + i] * rinv;
            Hpre_s[t * 4 + i]  = 1.0f / (1.0f + __expf(-(ap * pp + b_pre[i])));
            float pq = p_s[t * 32 + 4 + i] * rinv;
            Hpost_s[t * 4 + i] = 2.0f / (1.0f + __expf(-(aq * pq + b_post[i])));
        }
        #pragma unroll
        for (int i = 0; i < 16; ++i) {
            float tr = ar * (p_s[t * 32 + 8 + i] * rinv) + b_res[i];
            m[i] = __expf(fminf(tr, 20.0f));
        }
        #pragma unroll 1
        for (int it = 0; it < 10; ++it) {
            #pragma unroll
            for (int r = 0; r < 4; ++r) {
                float s = m[r*4] + m[r*4+1] + m[r*4+2] + m[r*4+3] + 1e-6f;
                float inv = 1.0f / s;
                m[r*4] *= inv; m[r*4+1] *= inv; m[r*4+2] *= inv; m[r*4+3] *= inv;
            }
            #pragma unroll
            for (int c = 0; c < 4; ++c) {
                float s = m[c] + m[4+c] + m[8+c] + m[12+c] + 1e-6f;
                float inv = 1.0f / s;
                m[c] *= inv; m[4+c] *= inv; m[8+c] *= inv; m[12+c] *= inv;
            }
        }
        #pragma unroll
        for (int i = 0; i < 16; ++i) M_s[t * 16 + i] = m[i];
    }
    __syncthreads();

    // ---- 5) aggregate + rmsnorm + mix + store (2 tokens per wave) ----
    for (int tt = 0; tt < 2; ++tt) {
        const int t = wave + tt * NWAVE;
        float hp0 = Hpre_s[t*4+0], hp1 = Hpre_s[t*4+1],
              hp2 = Hpre_s[t*4+2], hp3 = Hpre_s[t*4+3];
        const float* xb = xs + t * PITCH;

        float4 xa[8];
        float ss = 0.f;
        #pragma unroll
        for (int ci = 0; ci < 8; ++ci) {
            int col = (lane + 32 * ci) * 4;
            float4 v0 = *(const float4*)(xb + col);
            float4 v1 = *(const float4*)(xb + 1024 + col);
            float4 v2 = *(const float4*)(xb + 2048 + col);
            float4 v3 = *(const float4*)(xb + 3072 + col);
            float4 a;
            a.x = hp0*v0.x + hp1*v1.x + hp2*v2.x + hp3*v3.x;
            a.y = hp0*v0.y + hp1*v1.y + hp2*v2.y + hp3*v3.y;
            a.z = hp0*v0.z + hp1*v1.z + hp2*v2.z + hp3*v3.z;
            a.w = hp0*v0.w + hp1*v1.w + hp2*v2.w + hp3*v3.w;
            xa[ci] = a;
            ss += a.x*a.x + a.y*a.y + a.z*a.z + a.w*a.w;
        }
        for (int off = 16; off; off >>= 1) ss += __shfl_xor(ss, off, 32);
        const float rinv2 = rsqrtf(ss * (1.0f / 1024.0f) + 1e-6f);

        float hq[4], m[16];
        #pragma unroll
        for (int i = 0; i < 4; ++i)  hq[i] = Hpost_s[t * 4 + i];
        #pragma unroll
        for (int i = 0; i < 16; ++i) m[i]  = M_s[t * 16 + i];

        float* ob = out + ((tok0 + t) * 4) * (size_t)1024;
        #pragma unroll
        for (int ci = 0; ci < 8; ++ci) {
            int col = (lane + 32 * ci) * 4;
            float4 wv = *(const float4*)(w_s + col);
            float4 y;
            y.x = xa[ci].x * rinv2 * wv.x;
            y.y = xa[ci].y * rinv2 * wv.y;
            y.z = xa[ci].z * rinv2 * wv.z;
            y.w = xa[ci].w * rinv2 * wv.w;
            float4 v0 = *(const float4*)(xb + col);
            float4 v1 = *(const float4*)(xb + 1024 + col);
            float4 v2 = *(const float4*)(xb + 2048 + col);
            float4 v3 = *(const float4*)(xb + 3072 + col);
            #pragma unroll
            for (int i = 0; i < 4; ++i) {
                float4 o;
                o.x = hq[i]*y.x + m[i*4]*v0.x + m[i*4+1]*v1.x + m[i*4+2]*v2.x + m[i*4+3]*v3.x;
                o.y = hq[i]*y.y + m[i*4]*v0.y + m[i*4+1]*v1.y + m[i*4+2]*v2.y + m[i*4+3]*v3.y;
                o.z = hq[i]*y.z + m[i*4]*v0.z + m[i*4+1]*v1.z + m[i*4+2]*v2.z + m[i*4+3]*v3.z;
                o.w = hq[i]*y.w + m[i*4]*v0.w + m[i*4+1]*v1.w + m[i*4+2]*v2.w + m[i*4+3]*v3.w;
                *(float4*)(ob + i * (size_t)1024 + col) = o;
            }
        }
    }
}

extern "C" void kernel_launch(void* const* d_in, const int* in_sizes, int n_in,
                              void* d_out, int out_size, void* d_ws, size_t ws_size,
                              hipStream_t stream) {
    (void)in_sizes; (void)n_in; (void)out_size; (void)ws_size;
    const float* x        = (const float*)d_in[0];
    const float* phi_pre  = (const float*)d_in[1];
    const float* phi_post = (const float*)d_in[2];
    const float* phi_res  = (const float*)d_in[3];
    const float* b_pre    = (const float*)d_in[4];
    const float* b_post   = (const float*)d_in[5];
    const float* b_res    = (const float*)d_in[6];
    const float* a_pre    = (const float*)d_in[7];
    const float* a_post   = (const float*)d_in[8];
    const float* a_res    = (const float*)d_in[9];
    const float* rms_w    = (const float*)d_in[10];

    v2f* Bbuf = (v2f*)d_ws;   // 2 * 1024 * 32 float2 = 512 KB (L2-resident)

    build_Bbuf<<<256, 256, 0, stream>>>(phi_pre, phi_post, phi_res, Bbuf);

    const int T = 4 * 2048;
    const int nblocks = T / TOKENS_PER_WG;          // 512
    const size_t shbytes = LDS_FLOATS * sizeof(float); // ~286.5 KB < 320 KB
    mhc_fused<<<nblocks, 256, shbytes, stream>>>(
        x, Bbuf, b_pre, b_post, b_res, a_pre, a_post, a_res, rms_w,
        (float*)d_out);
}